// QNN_39676907888736
// MI455X (gfx1250) — compile-verified
//
#include <hip/hip_runtime.h>
#include <math.h>

typedef float v2f __attribute__((ext_vector_type(2)));
typedef float v8f __attribute__((ext_vector_type(8)));

#define NQ 4
#define DIM 16

// ---------------------------------------------------------------------------
// Setup kernel (1 thread): build Re(U), Im(U), W4 = ZSIGNS@fc_w^T (+ones), bias
// ws float layout: [0,256) Ure row-major [n][i], [256,512) Uim,
//                  [512,576) W4[16][4] (col 3 == 1.0), [576,579) fc_b
// ---------------------------------------------------------------------------
__device__ __forceinline__ void cmul2x2(const float* ar, const float* ai,
                                        const float* br, const float* bi,
                                        float* cr, float* ci) {
  for (int r = 0; r < 2; ++r)
    for (int c = 0; c < 2; ++c) {
      float sr = 0.f, si = 0.f;
      for (int k = 0; k < 2; ++k) {
        float xr = ar[r * 2 + k], xi = ai[r * 2 + k];
        float yr = br[k * 2 + c], yi = bi[k * 2 + c];
        sr += xr * yr - xi * yi;
        si += xr * yi + xi * yr;
      }
      cr[r * 2 + c] = sr;
      ci[r * 2 + c] = si;
    }
}

__global__ void qnn_setup_kernel(const float* __restrict__ params,
                                 const float* __restrict__ fc_w,
                                 const float* __restrict__ fc_b,
                                 float* __restrict__ ws) {
  if (threadIdx.x != 0 || blockIdx.x != 0) return;

  float gre[NQ][4], gim[NQ][4];
  for (int i = 0; i < NQ; ++i) {
    float hx = 0.5f * params[i * 3 + 0];
    float hy = 0.5f * params[i * 3 + 1];
    float hz = 0.5f * params[i * 3 + 2];
    float cx = cosf(hx), sx = sinf(hx);
    float cy = cosf(hy), sy = sinf(hy);
    float cz = cosf(hz), sz = sinf(hz);
    float axr[4] = {cx, 0.f, 0.f, cx},  axi[4] = {0.f, -sx, -sx, 0.f};  // RX
    float ayr[4] = {cy, -sy, sy, cy},   ayi[4] = {0.f, 0.f, 0.f, 0.f}; // RY
    float azr[4] = {cz, 0.f, 0.f, cz},  azi[4] = {-sz, 0.f, 0.f, sz};  // RZ
    float tr4[4], ti4[4];
    cmul2x2(ayr, ayi, axr, axi, tr4, ti4);        // RY @ RX
    cmul2x2(azr, azi, tr4, ti4, gre[i], gim[i]);  // RZ @ (RY@RX)
  }

  float re[DIM * DIM], im[DIM * DIM], tre[DIM * DIM], tim[DIM * DIM];
  re[0] = 1.f; im[0] = 0.f;
  int s = 1;
  for (int q = 0; q < NQ; ++q) {  // U = kron(kron(kron(U0,U1),U2),U3)
    int ns = s * 2;
    for (int a = 0; a < s; ++a)
      for (int b = 0; b < s; ++b) {
        float ar = re[a * s + b], ai = im[a * s + b];
        for (int u = 0; u < 2; ++u)
          for (int v = 0; v < 2; ++v) {
            float br = gre[q][u * 2 + v], bi = gim[q][u * 2 + v];
            tre[(a * 2 + u) * ns + (b * 2 + v)] = ar * br - ai * bi;
            tim[(a * 2 + u) * ns + (b * 2 + v)] = ar * bi + ai * br;
          }
      }
    s = ns;
    for (int k = 0; k < s * s; ++k) { re[k] = tre[k]; im[k] = tim[k]; }
  }

  // CNOT chain = row permutations
  for (int g = 0; g < NQ - 1; ++g) {
    int cbit = 1 << (NQ - 1 - g), tbit = 1 << (NQ - 1 - (g + 1));
    for (int k = 0; k < DIM; ++k) {
      int src = (k & cbit) ? (k ^ tbit) : k;
      for (int i = 0; i < DIM; ++i) {
        tre[k * DIM + i] = re[src * DIM + i];
        tim[k * DIM + i] = im[src * DIM + i];
      }
    }
    for (int k = 0; k < DIM * DIM; ++k) { re[k] = tre[k]; im[k] = tim[k]; }
  }

  for (int k = 0; k < DIM * DIM; ++k) { ws[k] = re[k]; ws[256 + k] = im[k]; }

  for (int n = 0; n < DIM; ++n) {
    for (int c = 0; c < 3; ++c) {
      float acc = 0.f;
      for (int w = 0; w < NQ; ++w) {
        float zs = 1.f - 2.f * (float)((n >> (NQ - 1 - w)) & 1);
        acc += zs * fc_w[c * NQ + w];
      }
      ws[512 + n * 4 + c] = acc;
    }
    ws[512 + n * 4 + 3] = 1.f;  // norm accumulator weight
  }
  for (int c = 0; c < 3; ++c) ws[576 + c] = fc_b[c];
}

// Swap values between lanes L and L+16 (v_permlanex16_b32, identity selects).
__device__ __forceinline__ float swap_half16(float v) {
  int i = __float_as_int(v);
  int r = __builtin_amdgcn_permlanex16(i, i, 0x76543210, 0xfedcba98, false, false);
  return __int_as_float(r);
}

// ---------------------------------------------------------------------------
// Main kernel: one wave32 processes 16 samples per tile via 8x
// V_WMMA_F32_16X16X4_F32:  D[n_out, sample] = U * X^T  (re and im).
// Reduction done in packed float2 form to map to v_pk_fma_f32.
// ---------------------------------------------------------------------------
__global__ __launch_bounds__(256) void qnn_main_kernel(
    const float* __restrict__ x, const float* __restrict__ ws,
    float* __restrict__ out, int ntiles) {
  const int lane = threadIdx.x & 31;
  const int half = lane >> 4;        // 0: K/M low half, 1: high half
  const int lrow = lane & 15;        // sample-in-tile (B/D column)
  const int wave = threadIdx.x >> 5;
  const int gwave = blockIdx.x * 8 + wave;
  const int nwaves = gridDim.x * 8;

  const float* __restrict__ Ure = ws;
  const float* __restrict__ Uim = ws + 256;
  const float* __restrict__ W4  = ws + 512;
  const float* __restrict__ bia = ws + 576;

  // Constant A operands: chunk c holds A[m=lrow][k = 4c + 2*half + {0,1}]
  v2f a_re[4], a_im[4];
#pragma unroll
  for (int c = 0; c < 4; ++c) {
    int off = lrow * 16 + 4 * c + 2 * half;
    a_re[c] = *(const v2f*)(Ure + off);
    a_im[c] = *(const v2f*)(Uim + off);
  }
  // This lane's reduction weights as row pairs: rows n = 2*rp + {0,1} + 8*half
  v2f w0p[4], w1p[4], w2p[4];
#pragma unroll
  for (int rp = 0; rp < 4; ++rp) {
    const float* pa = W4 + (2 * rp + 0 + 8 * half) * 4;
    const float* pb = W4 + (2 * rp + 1 + 8 * half) * 4;
    w0p[rp].x = pa[0]; w0p[rp].y = pb[0];
    w1p[rp].x = pa[1]; w1p[rp].y = pb[1];
    w2p[rp].x = pa[2]; w2p[rp].y = pb[2];
  }
  const float b0 = bia[0], b1 = bia[1], b2 = bia[2];

  for (int t = gwave; t < ntiles; t += nwaves) {
    const float* xt = x + (size_t)t * 256;  // 16 samples x 16 features
    // Prefetch next tile for this wave (global_prefetch_b8), lane spread
    __builtin_prefetch(xt + (size_t)nwaves * 256 + lane * 16, 0, 0);

    // B operand: X^T[k][sample], chunk c: k = 4c + 2*half + {0,1}
    v2f bch[4];
#pragma unroll
    for (int c = 0; c < 4; ++c)
      bch[c] = *(const v2f*)(xt + lrow * 16 + 4 * c + 2 * half);

    v8f accR = {}, accI = {};
#pragma unroll
    for (int c = 0; c < 4; ++c) {
      accR = __builtin_amdgcn_wmma_f32_16x16x4_f32(
          false, a_re[c], false, bch[c], (short)0, accR, false, false);
      accI = __builtin_amdgcn_wmma_f32_16x16x4_f32(
          false, a_im[c], false, bch[c], (short)0, accI, false, false);
    }

    // probs[n = 2rp+{0,1} + 8*half] for sample lrow; packed contraction
    v2f acc0 = {0.f, 0.f}, acc1 = {0.f, 0.f}, acc2 = {0.f, 0.f}, acc3 = {0.f, 0.f};
#pragma unroll
    for (int rp = 0; rp < 4; ++rp) {
      v2f vr; vr.x = accR[2 * rp]; vr.y = accR[2 * rp + 1];
      v2f vi; vi.x = accI[2 * rp]; vi.y = accI[2 * rp + 1];
      v2f p = vr * vr + vi * vi;     // v_pk_mul_f32 + v_pk_fma_f32
      acc0 += p * w0p[rp];           // v_pk_fma_f32
      acc1 += p * w1p[rp];
      acc2 += p * w2p[rp];
      acc3 += p;                     // v_pk_add_f32 (norm)
    }
    float s0 = acc0.x + acc0.y;
    float s1 = acc1.x + acc1.y;
    float s2 = acc2.x + acc2.y;
    float s3 = acc3.x + acc3.y;

    // merge the two n-halves (lanes L and L+16 hold same sample): VALU-only
    s0 += swap_half16(s0);
    s1 += swap_half16(s1);
    s2 += swap_half16(s2);
    s3 += swap_half16(s3);

    const float rinv = __builtin_amdgcn_rcpf(s3);  // 1/||x||^2 (unitarity)
    const size_t samp = (size_t)t * 16 + lrow;
    float* op = out + samp * 3;
    if (half == 0) {
      op[0] = s0 * rinv + b0;
      op[1] = s1 * rinv + b1;
    } else {
      op[2] = s2 * rinv + b2;
    }
  }
}

extern "C" void kernel_launch(void* const* d_in, const int* in_sizes, int n_in,
                              void* d_out, int out_size, void* d_ws, size_t ws_size,
                              hipStream_t stream) {
  const float* x      = (const float*)d_in[0];
  const float* params = (const float*)d_in[1];
  const float* fc_w   = (const float*)d_in[2];
  const float* fc_b   = (const float*)d_in[3];
  float* out = (float*)d_out;
  float* ws  = (float*)d_ws;

  const int ntiles = in_sizes[0] / 256;  // (B*16 floats) / (16 samples * 16)

  qnn_setup_kernel<<<1, 32, 0, stream>>>(params, fc_w, fc_b, ws);

  int blocks = 2048;                      // 8 wave32 per block, grid-stride
  int max_blocks = (ntiles + 7) / 8;
  if (blocks > max_blocks) blocks = max_blocks;
  if (blocks < 1) blocks = 1;
  qnn_main_kernel<<<blocks, 256, 0, stream>>>(x, ws, out, ntiles);
}